// RigBundleAdjustmentModel_74629351735836
// MI455X (gfx1250) — compile-verified
//
#include <hip/hip_runtime.h>
#include <stdint.h>

// 2048 cams * 48 B = 98304 B of LDS; CDNA5 WGP has 320 KB -> 3 blocks/WGP.
#define MAX_LDS_CAMS 2048

typedef unsigned int u32x4 __attribute__((ext_vector_type(4)));
typedef int          i32x8 __attribute__((ext_vector_type(8)));
typedef int          i32x4 __attribute__((ext_vector_type(4)));
typedef float        v2f   __attribute__((ext_vector_type(2)));
typedef float        v8f   __attribute__((ext_vector_type(8)));

// ---------------------------------------------------------------------------
// Per-camera math helpers
// ---------------------------------------------------------------------------
__device__ __forceinline__ void quat_to_rot(float x, float y, float z, float w,
                                            float R[9]) {
    float s   = x * x + y * y + z * z + w * w;
    float inv = 1.0f / sqrtf(s);
    x *= inv; y *= inv; z *= inv; w *= inv;
    R[0] = 1.0f - 2.0f * (y * y + z * z);
    R[1] = 2.0f * (x * y - w * z);
    R[2] = 2.0f * (x * z + w * y);
    R[3] = 2.0f * (x * y + w * z);
    R[4] = 1.0f - 2.0f * (x * x + z * z);
    R[5] = 2.0f * (y * z - w * x);
    R[6] = 2.0f * (x * z - w * y);
    R[7] = 2.0f * (y * z + w * x);
    R[8] = 1.0f - 2.0f * (x * x + y * y);
}

__device__ __forceinline__ void compute_cam_MV(int c,
                                               const float* __restrict__ base_poses,
                                               const float* __restrict__ rel_poses,
                                               const float* __restrict__ intrinsics,
                                               const int* __restrict__ lookup,
                                               float M[9], float v[3]) {
    const int bpi = lookup[2 * c + 0];
    const int rpi = lookup[2 * c + 1];

    const float* bp = base_poses + 7 * bpi;
    const float tb0 = bp[0], tb1 = bp[1], tb2 = bp[2];
    float Rb[9];
    quat_to_rot(bp[3], bp[4], bp[5], bp[6], Rb);

    float Rf[9], tf[3];
    if (rpi >= 0) {
        const float* rp = rel_poses + 7 * rpi;
        const float tr0 = rp[0], tr1 = rp[1], tr2 = rp[2];
        float Rr[9];
        quat_to_rot(rp[3], rp[4], rp[5], rp[6], Rr);
#pragma unroll
        for (int i = 0; i < 3; ++i)
#pragma unroll
            for (int j = 0; j < 3; ++j)
                Rf[3 * i + j] = Rb[3 * i + 0] * Rr[0 + j] +
                                Rb[3 * i + 1] * Rr[3 + j] +
                                Rb[3 * i + 2] * Rr[6 + j];
        tf[0] = tb0 + Rb[0] * tr0 + Rb[1] * tr1 + Rb[2] * tr2;
        tf[1] = tb1 + Rb[3] * tr0 + Rb[4] * tr1 + Rb[5] * tr2;
        tf[2] = tb2 + Rb[6] * tr0 + Rb[7] * tr1 + Rb[8] * tr2;
    } else {
#pragma unroll
        for (int k = 0; k < 9; ++k) Rf[k] = Rb[k];
        tf[0] = tb0; tf[1] = tb1; tf[2] = tb2;
    }

    const float* K = intrinsics + 9 * c;
#pragma unroll
    for (int i = 0; i < 3; ++i) {
#pragma unroll
        for (int j = 0; j < 3; ++j)
            M[3 * i + j] = K[3 * i + 0] * Rf[0 + j] +
                           K[3 * i + 1] * Rf[3 + j] +
                           K[3 * i + 2] * Rf[6 + j];
        v[i] = K[3 * i + 0] * tf[0] + K[3 * i + 1] * tf[1] + K[3 * i + 2] * tf[2];
    }
}

// ---------------------------------------------------------------------------
// Kernel 1: one wave per camera. The 4x4 homogeneous pose composition
// T_comp = T_base * T_rel is done with V_WMMA_F32_16X16X4_F32 (16x4 . 4x16):
//   A rows 0-3  = T_base   (lanes 0-15 hold K={0,1}, lanes 16-31 K={2,3})
//   B cols 0-3  = T_rel    (row striped across lanes; V0 rows {0,2}, V1 {1,3})
//   T_rel = I when the camera has no rig offset (exact under fp multiply).
// After the WMMA, column j of T_comp is lane-local (D VGPRs 0-3 on lane j),
// so lanes 0-3 fold the intrinsics: camMV[c] = {K*R_f (9), K*t_f (3)}.
// ---------------------------------------------------------------------------
__global__ __launch_bounds__(256) void precompute_cam_wave_kernel(
        const float* __restrict__ base_poses,
        const float* __restrict__ rel_poses,
        const float* __restrict__ intrinsics,
        const int* __restrict__ lookup,
        float* __restrict__ camMV,
        int C) {
    const int wave = (blockIdx.x * blockDim.x + threadIdx.x) >> 5;  // wave32
    const int lane = threadIdx.x & 31;
    if (wave >= C) return;  // wave-uniform: EXEC stays all-1s for the WMMA
    const int c = wave;

#if __has_builtin(__builtin_amdgcn_wmma_f32_16x16x4_f32)
    const int bpi = lookup[2 * c + 0];
    const int rpi = lookup[2 * c + 1];

    const float* bp = base_poses + 7 * bpi;
    float Rb[9];
    quat_to_rot(bp[3], bp[4], bp[5], bp[6], Rb);
    const float tb[3] = {bp[0], bp[1], bp[2]};

    float Rr[9] = {1.f, 0.f, 0.f, 0.f, 1.f, 0.f, 0.f, 0.f, 1.f};
    float tr[3] = {0.f, 0.f, 0.f};
    if (rpi >= 0) {
        const float* rp = rel_poses + 7 * rpi;
        quat_to_rot(rp[3], rp[4], rp[5], rp[6], Rr);
        tr[0] = rp[0]; tr[1] = rp[1]; tr[2] = rp[2];
    }

    // Homogeneous transform elements (row r, col k), r,k in [0,4)
    auto Tb = [&](int r, int k) -> float {
        if (r < 3) return (k < 3) ? Rb[3 * r + k] : tb[r];
        return (k == 3) ? 1.0f : 0.0f;
    };
    auto Tr = [&](int r, int k) -> float {
        if (r < 3) return (k < 3) ? Rr[3 * r + k] : tr[r];
        return (k == 3) ? 1.0f : 0.0f;
    };

    const int half = lane >> 4;  // 0: lanes 0-15 (K 0,1) | 1: lanes 16-31 (K 2,3)
    const int idx  = lane & 15;

    // A operand (16x4 f32): lane holds A[row=idx][K=2*half+{0,1}]
    v2f A;
    A.x = (idx < 4) ? Tb(idx, 2 * half + 0) : 0.0f;
    A.y = (idx < 4) ? Tb(idx, 2 * half + 1) : 0.0f;

    // B operand (4x16 f32): lane holds B[K=2*half+{0,1}][col=idx]
    v2f Bv;
    Bv.x = (idx < 4) ? Tr(2 * half + 0, idx) : 0.0f;
    Bv.y = (idx < 4) ? Tr(2 * half + 1, idx) : 0.0f;

    v8f Cacc = {0.f, 0.f, 0.f, 0.f, 0.f, 0.f, 0.f, 0.f};
    // 8 args: (neg_a, A, neg_b, B, c_mod, C, reuse_a, reuse_b)
    v8f D = __builtin_amdgcn_wmma_f32_16x16x4_f32(
        false, A, false, Bv, (short)0, Cacc, false, false);

    // D[r] on lane j (j<16) = T_comp[r][j] for r<8 -> column j is lane-local.
    if (lane < 4) {
        const float T0 = D[0], T1 = D[1], T2 = D[2];
        const float* K = intrinsics + 9 * c;
        const float r0 = fmaf(K[0], T0, fmaf(K[1], T1, K[2] * T2));
        const float r1 = fmaf(K[3], T0, fmaf(K[4], T1, K[5] * T2));
        const float r2 = fmaf(K[6], T0, fmaf(K[7], T1, K[8] * T2));
        float* o = camMV + 12 * c;
        if (lane < 3) {             // column j of M = K*R_f
            o[lane + 0] = r0;
            o[lane + 3] = r1;
            o[lane + 6] = r2;
        } else {                    // column 3 -> v = K*t_f
            o[9]  = r0;
            o[10] = r1;
            o[11] = r2;
        }
    }
#else
    if (lane == 0) {
        float M[9], v[3];
        compute_cam_MV(c, base_poses, rel_poses, intrinsics, lookup, M, v);
        float* o = camMV + 12 * c;
#pragma unroll
        for (int k = 0; k < 9; ++k) o[k] = M[k];
        o[9] = v[0]; o[10] = v[1]; o[11] = v[2];
    }
#endif
}

// ---------------------------------------------------------------------------
// Streaming residual body.
// cam layout per camera (3 x float4): [M00 M01 M02 M10 | M11 M12 M20 M21 | M22 v0 v1 v2]
// ---------------------------------------------------------------------------
__device__ __forceinline__ void residual_body(long long i,
                                              const float4* __restrict__ obs,
                                              const float* __restrict__ points,
                                              const float4* cam,
                                              float2* __restrict__ out) {
    const float4 o = obs[i];
    const int c = (int)o.x;
    const int p = (int)o.y;
    const float4 m0 = cam[3 * c + 0];
    const float4 m1 = cam[3 * c + 1];
    const float4 m2 = cam[3 * c + 2];
    const float* pp = points + 3 * (long long)p;
    const float px = pp[0], py = pp[1], pz = pp[2];
    const float X = fmaf(m0.x, px, fmaf(m0.y, py, fmaf(m0.z, pz, m2.y)));
    const float Y = fmaf(m0.w, px, fmaf(m1.x, py, fmaf(m1.y, pz, m2.z)));
    const float Z = fmaf(m1.z, px, fmaf(m1.w, py, fmaf(m2.x, pz, m2.w)));
    const float invZ = 1.0f / Z;
    out[i] = make_float2(fmaf(X, invZ, -o.z), fmaf(Y, invZ, -o.w));
}

// ---------------------------------------------------------------------------
// Kernel 2: main streaming kernel. Stages the camera table into LDS with the
// CDNA5 Tensor Data Mover (TENSORcnt) when available, else per-lane
// global_load_async_to_lds_b128 (ASYNCcnt); then grid-strides over obs.
// ---------------------------------------------------------------------------
__global__ __launch_bounds__(256) void rig_residual_kernel(
        const float4* __restrict__ obs,
        const float* __restrict__ points,
        const float* __restrict__ camMV,
        float2* __restrict__ out,
        long long n, int C) {
    __shared__ float4 s_cam[MAX_LDS_CAMS * 3];

    const long long stride = (long long)gridDim.x * blockDim.x;
    const long long i0     = (long long)blockIdx.x * blockDim.x + threadIdx.x;

    if (C <= MAX_LDS_CAMS) {
        const uint32_t ldsAddr = (uint32_t)(uintptr_t)(&s_cam[0]);  // flat low 32 = LDS offset

#if __has_builtin(__builtin_amdgcn_tensor_load_to_lds) && \
    __has_builtin(__builtin_amdgcn_s_wait_tensorcnt)
        // --- TDM path: one descriptor moves the whole C*48-byte table ---
        if (threadIdx.x == 0) {
            const uint64_t ga     = (uint64_t)(uintptr_t)camMV;
            const uint32_t nElems = (uint32_t)C * 6u;  // 8-byte elements

            // D# group 0: count=1 | lds_addr | global_addr[31:0] |
            //             global_addr[56:32] + type=2 ("image")
            u32x4 g0;
            g0[0] = 1u;
            g0[1] = ldsAddr;
            g0[2] = (uint32_t)ga;
            g0[3] = (uint32_t)((ga >> 32) & 0x01FFFFFFu) | (2u << 30);

            // D# group 1: 1-D tile of nElems 8-byte elements.
            i32x8 g1;
            g1[0] = (int)(3u << 16);                        // data_size=8B, mask=0
            g1[1] = (int)((nElems & 0xFFFFu) << 16);        // tensor_dim0[15:0]
            g1[2] = (int)((nElems >> 16) | (1u << 16));     // tensor_dim0[31:16] | tensor_dim1=1
            g1[3] = (int)((nElems & 0xFFFFu) << 16);        // tensor_dim1 hi=0 | tile_dim0
            g1[4] = 0;                                      // tile_dim1=0, tile_dim2=0
            g1[5] = (int)nElems;                            // tensor_dim0_stride[31:0]
            g1[6] = 0;                                      // stride hi | dim1_stride lo
            g1[7] = 0;

            i32x4 g2 = {0, 0, 0, 0};
            i32x4 g3 = {0, 0, 0, 0};
            i32x8 g4 = {0, 0, 0, 0, 0, 0, 0, 0};
            // clang-23 / therock 6-arg form: (g0, g1, g2, g3, extra, cpol)
            __builtin_amdgcn_tensor_load_to_lds(g0, g1, g2, g3, g4, 0);
            __builtin_amdgcn_s_wait_tensorcnt(0);
        }
#else
        // --- fallback: per-lane async copies, ASYNCcnt-tracked ---
        const uint32_t total = (uint32_t)C * 48u;  // multiple of 16
        const char*    src   = (const char*)camMV;
        for (uint32_t off = (uint32_t)threadIdx.x * 16u; off < total;
             off += (uint32_t)blockDim.x * 16u) {
            const uint32_t dst = ldsAddr + off;
            const char*    g   = src + off;
            asm volatile("global_load_async_to_lds_b128 %0, %1, off"
                         :
                         : "v"(dst), "v"(g)
                         : "memory");
        }
#if __has_builtin(__builtin_amdgcn_s_wait_asynccnt)
        __builtin_amdgcn_s_wait_asynccnt(0);
#else
        asm volatile("s_wait_asynccnt 0" ::: "memory");
#endif
#endif
        __syncthreads();

        // 2-way grid-stride: two independent gather chains in flight per lane.
        long long i = i0;
        for (; i + stride < n; i += 2 * stride) {
            if (i + 2 * stride < n) __builtin_prefetch(&obs[i + 2 * stride], 0, 0);
            residual_body(i, obs, points, &s_cam[0], out);
            residual_body(i + stride, obs, points, &s_cam[0], out);
        }
        if (i < n) residual_body(i, obs, points, &s_cam[0], out);
    } else {
        // Camera table too big for LDS: gather straight from L2.
        for (long long i = i0; i < n; i += stride)
            residual_body(i, obs, points, (const float4*)camMV, out);
    }
}

// ---------------------------------------------------------------------------
// Fused fallback (used only if workspace can't hold the camera table).
// ---------------------------------------------------------------------------
__global__ __launch_bounds__(256) void rig_residual_fused_kernel(
        const float4* __restrict__ obs,
        const float* __restrict__ base_poses,
        const float* __restrict__ rel_poses,
        const float* __restrict__ points,
        const float* __restrict__ intrinsics,
        const int* __restrict__ lookup,
        float2* __restrict__ out,
        long long n) {
    const long long stride = (long long)gridDim.x * blockDim.x;
    for (long long i = (long long)blockIdx.x * blockDim.x + threadIdx.x; i < n;
         i += stride) {
        const float4 o = obs[i];
        const int c = (int)o.x;
        const int p = (int)o.y;
        float M[9], v[3];
        compute_cam_MV(c, base_poses, rel_poses, intrinsics, lookup, M, v);
        const float* pp = points + 3 * (long long)p;
        const float px = pp[0], py = pp[1], pz = pp[2];
        const float X = fmaf(M[0], px, fmaf(M[1], py, fmaf(M[2], pz, v[0])));
        const float Y = fmaf(M[3], px, fmaf(M[4], py, fmaf(M[5], pz, v[1])));
        const float Z = fmaf(M[6], px, fmaf(M[7], py, fmaf(M[8], pz, v[2])));
        const float invZ = 1.0f / Z;
        out[i] = make_float2(fmaf(X, invZ, -o.z), fmaf(Y, invZ, -o.w));
    }
}

// ---------------------------------------------------------------------------
// Launch
// ---------------------------------------------------------------------------
extern "C" void kernel_launch(void* const* d_in, const int* in_sizes, int n_in,
                              void* d_out, int out_size, void* d_ws, size_t ws_size,
                              hipStream_t stream) {
    const float* observations = (const float*)d_in[0];  // (N,4)
    const float* base_poses   = (const float*)d_in[1];  // (B,7)
    const float* rel_poses    = (const float*)d_in[2];  // (R,7)
    const float* points       = (const float*)d_in[3];  // (P,3)
    const float* intrinsics   = (const float*)d_in[4];  // (C,3,3)
    const int*   lookup       = (const int*)d_in[5];    // (C,2)

    const long long N = (long long)(in_sizes[0] / 4);
    const int       C = in_sizes[4] / 9;

    const size_t needWs = (size_t)C * 12u * sizeof(float);
    if (d_ws != nullptr && ws_size >= needWs) {
        float* camMV = (float*)d_ws;
        // One wave (32 threads) per camera.
        const long long pre_threads = (long long)C * 32;
        const int pre_blocks = (int)((pre_threads + 255) / 256);
        precompute_cam_wave_kernel<<<pre_blocks, 256, 0, stream>>>(
            base_poses, rel_poses, intrinsics, lookup, camMV, C);

        long long want = (N + 255) / 256;
        int blocks = (int)(want < 1024 ? want : 1024);
        if (blocks < 1) blocks = 1;
        rig_residual_kernel<<<blocks, 256, 0, stream>>>(
            (const float4*)observations, points, camMV, (float2*)d_out, N, C);
    } else {
        long long want = (N + 255) / 256;
        int blocks = (int)(want < 2048 ? want : 2048);
        if (blocks < 1) blocks = 1;
        rig_residual_fused_kernel<<<blocks, 256, 0, stream>>>(
            (const float4*)observations, base_poses, rel_poses, points,
            intrinsics, lookup, (float2*)d_out, N);
    }
}